// WeightedSGScheme_10428180594968
// MI455X (gfx1250) — compile-verified
//
#include <hip/hip_runtime.h>
#include <hip/hip_bf16.h>

typedef __attribute__((ext_vector_type(2))) float v2f;
typedef __attribute__((ext_vector_type(8))) float v8f;

#define NR 256   // n_recon rows used in the fit

// ---------------------------------------------------------------------------
// Phase 1: build Beff = (V^T W V)^{-1} V^T W  (4 x 256), pre-swizzled into the
// B-matrix lane layout of V_WMMA_F32_16X16X4_F32, zero-padded to N=16.
// Bws layout: step s (=K-chunk i0/4), lane l, j in {0,1}:
//   Bws[s*64 + l*2 + j] = B[k][n], k = (l<16 ? 0:2)+j, n = l&15
// ---------------------------------------------------------------------------
__global__ __launch_bounds__(256) void sg_phase1(const float* __restrict__ t_all,
                                                 const float* __restrict__ weights,
                                                 float* __restrict__ Bws)
{
    __shared__ float sG[16];
    __shared__ float sGinv[16];
    __shared__ float sAw[4 * NR];

    const int i = threadIdx.x;            // 0..255 == row index
    if (i < 16) sG[i] = 0.0f;
    __syncthreads();

    // Vandermonde row (normalized) and softplus weight
    const float tc = t_all[NR / 2];
    const float tr = t_all[i] - tc;
    const float wr = weights[i];
    const float w  = fmaxf(wr, 0.0f) + log1pf(expf(-fabsf(wr)));   // softplus
    float v[4];
    v[0] = 1.0f;
    v[1] = tr;                             // / max(1,1)
    v[2] = tr * tr * 0.5f;                 // / 2
    v[3] = tr * tr * tr * (1.0f / 3.0f);   // / 3

    #pragma unroll
    for (int p = 0; p < 4; ++p) sAw[p * NR + i] = v[p] * w;

    #pragma unroll
    for (int p = 0; p < 4; ++p)
        #pragma unroll
        for (int q = 0; q < 4; ++q)
            atomicAdd(&sG[p * 4 + q], v[p] * v[q] * w);   // ds_add_f32
    __syncthreads();

    if (i == 0) {
        // Gauss-Jordan inverse of the SPD 4x4 (no pivoting needed)
        float M[4][8];
        for (int r = 0; r < 4; ++r)
            for (int c = 0; c < 4; ++c) {
                M[r][c]     = sG[r * 4 + c];
                M[r][4 + c] = (r == c) ? 1.0f : 0.0f;
            }
        for (int col = 0; col < 4; ++col) {
            const float inv = 1.0f / M[col][col];
            for (int c = 0; c < 8; ++c) M[col][c] *= inv;
            for (int r = 0; r < 4; ++r) {
                if (r == col) continue;
                const float f = M[r][col];
                for (int c = 0; c < 8; ++c) M[r][c] -= f * M[col][c];
            }
        }
        for (int r = 0; r < 4; ++r)
            for (int c = 0; c < 4; ++c) sGinv[r * 4 + c] = M[r][4 + c];
    }
    __syncthreads();

    // Beff[n][i] = sum_q Ginv[n][q] * Aw[q][i]
    float beff[4];
    #pragma unroll
    for (int n = 0; n < 4; ++n) {
        float acc = 0.0f;
        #pragma unroll
        for (int q = 0; q < 4; ++q) acc += sGinv[n * 4 + q] * sAw[q * NR + i];
        beff[n] = acc;
    }

    // Scatter into the WMMA B lane layout (this thread owns K-index i)
    const int s = i >> 2;
    const int k = i & 3;
    const int j = k & 1;
    const int lbase = (k < 2) ? 0 : 16;
    #pragma unroll
    for (int n = 0; n < 16; ++n) {
        const float val = (n < 4) ? beff[n] : 0.0f;
        Bws[s * 64 + (lbase + n) * 2 + j] = val;
    }
}

// ---------------------------------------------------------------------------
// Phase 2: out^T(tile of 16 d's, 4 coeffs) = X^T_tile(16x256) * Beff^T(256x4)
// via 64 chained V_WMMA_F32_16X16X4_F32 per wave. Memory-bound: streams X once.
// ---------------------------------------------------------------------------
__global__ __launch_bounds__(256) void sg_phase2(const float* __restrict__ X,
                                                 const float* __restrict__ Bws,
                                                 float* __restrict__ out,
                                                 int Dc, int numTiles)
{
    const int lane = threadIdx.x & 31;
    const int wave = threadIdx.x >> 5;
    const int tile = blockIdx.x * 8 + wave;
    if (tile >= numTiles) return;          // wave-uniform: EXEC stays all-ones

    const long d0 = (long)tile * 16;
    const int  m  = lane & 15;             // A-matrix row M (= d offset)
    const int  hi = lane >> 4;             // 0: K={0,1}, 1: K={2,3}
    const int  kb = hi * 2;

    long col = d0 + m;
    if (col >= Dc) col = Dc - 1;           // clamp: row m only affects D row m
    const float* xp = X + (long)kb * Dc + col;

    v8f c = {0.0f, 0.0f, 0.0f, 0.0f, 0.0f, 0.0f, 0.0f, 0.0f};

    for (int s = 0; s < NR / 4; ++s) {
        const float* xs = xp + (long)s * 4 * Dc;
        v2f a;
        a.x = xs[0];                       // A[m][kb+0]
        a.y = xs[Dc];                      // A[m][kb+1]
        v2f b = *(const v2f*)(Bws + s * 64 + lane * 2);
        __builtin_prefetch(xs + 8 * (long)Dc, 0, 0);   // global_prefetch_b8
        // 8 args: (neg_a, A, neg_b, B, c_mod, C, reuse_a, reuse_b)
        c = __builtin_amdgcn_wmma_f32_16x16x4_f32(
                false, a, false, b, (short)0, c, false, false);
    }

    // C/D layout: VGPR r -> (M=r, lanes 0-15) / (M=r+8, lanes 16-31); N = lane&15
    const long dbase = d0 + (long)hi * 8;
    if (d0 + 16 <= (long)Dc) {
        // Full tile (always true when D % 16 == 0): branch-free predicated stores
        if (m < 4) {
            #pragma unroll
            for (int r = 0; r < 8; ++r) {
                out[(dbase + r) * 4 + m] = c[r];
            }
        }
    } else {
        // Ragged last tile (dead at D = 100000, kept for generality)
        if (m < 4) {
            #pragma unroll
            for (int r = 0; r < 8; ++r) {
                const long d = dbase + r;
                if (d < Dc) out[d * 4 + m] = c[r];
            }
        }
    }
}

extern "C" void kernel_launch(void* const* d_in, const int* in_sizes, int n_in,
                              void* d_out, int out_size, void* d_ws, size_t ws_size,
                              hipStream_t stream) {
    const float* t_all   = (const float*)d_in[0];
    const float* x_all   = (const float*)d_in[1];
    const float* weights = (const float*)d_in[2];
    (void)d_in[3]; (void)n_in; (void)out_size; (void)ws_size;

    const int T  = in_sizes[0];
    const int Dc = in_sizes[1] / T;        // 100000

    float* Bws = (float*)d_ws;             // 64 steps * 64 floats = 16 KB

    hipLaunchKernelGGL(sg_phase1, dim3(1), dim3(256), 0, stream,
                       t_all, weights, Bws);

    const int numTiles = (Dc + 15) / 16;   // 6250
    const int blocks   = (numTiles + 7) / 8;
    hipLaunchKernelGGL(sg_phase2, dim3(blocks), dim3(256), 0, stream,
                       x_all, Bws, (float*)d_out, Dc, numTiles);
}